// CT_loss_65816078844169
// MI455X (gfx1250) — compile-verified
//
#include <hip/hip_runtime.h>

// Problem constants (from reference setup): B=64, H=W=128
#define HW          16384
#define NBATCH      64
#define CHUNKS      4                    // blocks per batch
#define NBLOCKS     (NBATCH * CHUNKS)    // 256 stage-1 blocks
#define NTHREADS    256                  // 8 wave32 waves
#define WS_STRIDE   16                   // floats per partial row (6 used, padded to 16 for WMMA)

typedef __attribute__((ext_vector_type(2))) float v2f;
typedef __attribute__((ext_vector_type(8))) float v8f;

// ---------------------------------------------------------------------------
// Stage 1: streaming pass. Each block = one (batch, quarter-of-image) chunk.
// float4 loads (global_load_b128), all rotation math in registers,
// deterministic wave32 shfl_xor reduction + LDS cross-wave combine.
// Writes 16-float partial row per block: [s0,s1,s2,m0,m1,m2, 0...0]
// ---------------------------------------------------------------------------
__global__ __launch_bounds__(NTHREADS) void ct_stage1(
    const float* __restrict__ rots,   // (B,3,3)
    const float* __restrict__ P0,     // (B,3,H,W)
    const float* __restrict__ Q0,     // (B,6,H,W)
    const float* __restrict__ occ,    // (B,3,H,W)
    const float* __restrict__ ext,    // (B,3)
    const float* __restrict__ trans,  // (B,3)
    float* __restrict__ ws)
{
    const int blk   = blockIdx.x;
    const int b     = blk >> 2;       // batch
    const int chunk = blk & 3;        // quarter of the image
    const int tid   = threadIdx.x;

    // Per-batch constants (uniform -> scalar loads)
    const float R00 = rots[b*9+0], R01 = rots[b*9+1], R02 = rots[b*9+2];
    const float R10 = rots[b*9+3], R11 = rots[b*9+4], R12 = rots[b*9+5];
    const float R20 = rots[b*9+6], R21 = rots[b*9+7], R22 = rots[b*9+8];
    const float t0  = trans[b*3+0], t1 = trans[b*3+1], t2 = trans[b*3+2];
    const float e0  = ext[b*3+0],   e1 = ext[b*3+1],   e2 = ext[b*3+2];
    // RTt[a] = sum_i R[i][a] * t[i]
    const float RTt0 = R00*t0 + R10*t1 + R20*t2;
    const float RTt1 = R01*t0 + R11*t1 + R21*t2;
    const float RTt2 = R02*t0 + R12*t1 + R22*t2;

    const float4* Pc0 = (const float4*)(P0 + (size_t)b*3*HW + 0*HW);
    const float4* Pc1 = (const float4*)(P0 + (size_t)b*3*HW + 1*HW);
    const float4* Pc2 = (const float4*)(P0 + (size_t)b*3*HW + 2*HW);
    const float4* Qc0 = (const float4*)(Q0 + (size_t)b*6*HW + 0*HW);
    const float4* Qc1 = (const float4*)(Q0 + (size_t)b*6*HW + 1*HW);
    const float4* Qc2 = (const float4*)(Q0 + (size_t)b*6*HW + 2*HW);
    const float4* Qc3 = (const float4*)(Q0 + (size_t)b*6*HW + 3*HW);
    const float4* Qc4 = (const float4*)(Q0 + (size_t)b*6*HW + 4*HW);
    const float4* Qc5 = (const float4*)(Q0 + (size_t)b*6*HW + 5*HW);
    const float4* Mc0 = (const float4*)(occ + (size_t)b*3*HW + 0*HW);
    const float4* Mc1 = (const float4*)(occ + (size_t)b*3*HW + 1*HW);
    const float4* Mc2 = (const float4*)(occ + (size_t)b*3*HW + 2*HW);

    float s0 = 0.f, s1 = 0.f, s2 = 0.f;
    float m0 = 0.f, m1 = 0.f, m2 = 0.f;

    #pragma unroll
    for (int it = 0; it < 4; ++it) {
        // float4 index within this image: chunk*(4096/4 per chunk)... hw/4 = 4096
        const int n4 = chunk * 1024 + it * NTHREADS + tid;

        const float4 vp0 = Pc0[n4], vp1 = Pc1[n4], vp2 = Pc2[n4];
        const float4 vq0 = Qc0[n4], vq1 = Qc1[n4], vq2 = Qc2[n4];
        const float4 vq3 = Qc3[n4], vq4 = Qc4[n4], vq5 = Qc5[n4];
        const float4 vm0 = Mc0[n4], vm1 = Mc1[n4], vm2 = Mc2[n4];

        #pragma unroll
        for (int u = 0; u < 4; ++u) {
            const float p0c = ((&vp0.x)[u] - 0.5f) * e0;
            const float p1c = ((&vp1.x)[u] - 0.5f) * e1;
            const float p2c = ((&vp2.x)[u] - 0.5f) * e2;
            // ext_q = [e1,e2, e0,e2, e0,e1]
            const float cx0 = ((&vq0.x)[u] - 0.5f) * e1;  // q0x = (0, cx0, cx1)
            const float cx1 = ((&vq1.x)[u] - 0.5f) * e2;
            const float cy0 = ((&vq2.x)[u] - 0.5f) * e0;  // q0y = (cy0, 0, cy1)
            const float cy1 = ((&vq3.x)[u] - 0.5f) * e2;
            const float cz0 = ((&vq4.x)[u] - 0.5f) * e0;  // q0z = (cz0, cz1, 0)
            const float cz1 = ((&vq5.x)[u] - 0.5f) * e1;

            const float RP0 = R00*p0c + R01*p1c + R02*p2c + t0;
            const float RP1 = R10*p0c + R11*p1c + R12*p2c + t1;
            const float RP2 = R20*p0c + R21*p1c + R22*p2c + t2;

            const float RX0 = R01*cx0 + R02*cx1 + t0;
            const float RX1 = R11*cx0 + R12*cx1 + t1;
            const float RX2 = R21*cx0 + R22*cx1 + t2;

            const float RY0 = R00*cy0 + R02*cy1 + t0;
            const float RY1 = R10*cy0 + R12*cy1 + t1;
            const float RY2 = R20*cy0 + R22*cy1 + t2;

            const float RZ0 = R00*cz0 + R01*cz1 + t0;
            const float RZ1 = R10*cz0 + R11*cz1 + t1;
            const float RZ2 = R20*cz0 + R21*cz1 + t2;

            // axis 0
            {
                const float proj = R00*RP0 + R10*RP1 + R20*RP2;
                const float d0 = RTt0*RP0 - proj*RX0;
                const float d1 = RTt0*RP1 - proj*RX1;
                const float d2 = RTt0*RP2 - proj*RX2;
                const float ma = (&vm0.x)[u];
                s0 += sqrtf(d0*d0 + d1*d1 + d2*d2) * ma;
                m0 += ma;
            }
            // axis 1
            {
                const float proj = R01*RP0 + R11*RP1 + R21*RP2;
                const float d0 = RTt1*RP0 - proj*RY0;
                const float d1 = RTt1*RP1 - proj*RY1;
                const float d2 = RTt1*RP2 - proj*RY2;
                const float ma = (&vm1.x)[u];
                s1 += sqrtf(d0*d0 + d1*d1 + d2*d2) * ma;
                m1 += ma;
            }
            // axis 2
            {
                const float proj = R02*RP0 + R12*RP1 + R22*RP2;
                const float d0 = RTt2*RP0 - proj*RZ0;
                const float d1 = RTt2*RP1 - proj*RZ1;
                const float d2 = RTt2*RP2 - proj*RZ2;
                const float ma = (&vm2.x)[u];
                s2 += sqrtf(d0*d0 + d1*d1 + d2*d2) * ma;
                m2 += ma;
            }
        }
    }

    // Deterministic wave32 reduction
    float vals[6] = {s0, s1, s2, m0, m1, m2};
    #pragma unroll
    for (int k = 0; k < 6; ++k) {
        #pragma unroll
        for (int off = 16; off > 0; off >>= 1)
            vals[k] += __shfl_xor(vals[k], off, 32);
    }

    __shared__ float red[8][6];
    const int wave = tid >> 5;
    const int lane = tid & 31;
    if (lane == 0) {
        #pragma unroll
        for (int k = 0; k < 6; ++k) red[wave][k] = vals[k];
    }
    __syncthreads();

    // Fixed-order cross-wave combine; write padded 16-float partial row
    if (tid < WS_STRIDE) {
        float v = 0.0f;
        if (tid < 6) {
            #pragma unroll
            for (int w = 0; w < 8; ++w) v += red[w][tid];
        }
        ws[blk * WS_STRIDE + tid] = v;
    }
}

// ---------------------------------------------------------------------------
// Stage 2: deterministic final reduction of the 256x16 partial matrix via
// V_WMMA_F32_16X16X4_F32: colsum = 1^T * P, chained over 64 WMMAs with 4
// independent accumulators. Lane j of D row 0 holds column-sum j.
// ---------------------------------------------------------------------------
__global__ __launch_bounds__(32) void ct_stage2(
    const float* __restrict__ ws, float* __restrict__ out)
{
    const int lane = threadIdx.x;          // 0..31, wave32
    const int col  = lane & 15;            // N column
    const int rsel = (lane < 16) ? 0 : 2;  // B layout: VGPR0 holds K=0 (lanes 0-15) / K=2 (lanes 16-31)

    v2f a; a[0] = 1.0f; a[1] = 1.0f;       // A = ones(16x4)
    v8f c0 = {}; v8f c1 = {}; v8f c2 = {}; v8f c3 = {};

    for (int i = 0; i < 16; ++i) {
        const int r = i * 16;              // 16 partial rows per iteration, 4 per WMMA
        v2f bm;
        bm[0] = ws[(r + 0 + rsel) * WS_STRIDE + col];
        bm[1] = ws[(r + 1 + rsel) * WS_STRIDE + col];
        c0 = __builtin_amdgcn_wmma_f32_16x16x4_f32(false, a, false, bm, (short)0, c0, false, false);
        bm[0] = ws[(r + 4 + rsel) * WS_STRIDE + col];
        bm[1] = ws[(r + 5 + rsel) * WS_STRIDE + col];
        c1 = __builtin_amdgcn_wmma_f32_16x16x4_f32(false, a, false, bm, (short)0, c1, false, false);
        bm[0] = ws[(r + 8 + rsel) * WS_STRIDE + col];
        bm[1] = ws[(r + 9 + rsel) * WS_STRIDE + col];
        c2 = __builtin_amdgcn_wmma_f32_16x16x4_f32(false, a, false, bm, (short)0, c2, false, false);
        bm[0] = ws[(r + 12 + rsel) * WS_STRIDE + col];
        bm[1] = ws[(r + 13 + rsel) * WS_STRIDE + col];
        c3 = __builtin_amdgcn_wmma_f32_16x16x4_f32(false, a, false, bm, (short)0, c3, false, false);
    }

    const v8f c = c0 + c1 + c2 + c3;       // elementwise merge of independent accumulators
    const float colsum = c[0];             // D row M=0: lane j (0..15) = column-sum j

    const float s0 = __shfl(colsum, 0, 32);
    const float s1 = __shfl(colsum, 1, 32);
    const float s2 = __shfl(colsum, 2, 32);
    const float g0 = __shfl(colsum, 3, 32);
    const float g1 = __shfl(colsum, 4, 32);
    const float g2 = __shfl(colsum, 5, 32);

    if (lane == 0) {
        const float b3 = 192.0f;           // B * 3
        float total = g0 + g1 + g2;
        total = fmaxf(total, 1.0f);
        float loss = 0.0f;
        if (g0 >= b3) loss += s0;          // jnp.where(mask.sum() < b3, 0, s)
        if (g1 >= b3) loss += s1;
        if (g2 >= b3) loss += s2;
        out[0] = loss / total;
    }
}

// ---------------------------------------------------------------------------
extern "C" void kernel_launch(void* const* d_in, const int* in_sizes, int n_in,
                              void* d_out, int out_size, void* d_ws, size_t ws_size,
                              hipStream_t stream) {
    (void)in_sizes; (void)n_in; (void)out_size; (void)ws_size;
    const float* rots  = (const float*)d_in[0];
    const float* P0    = (const float*)d_in[1];
    const float* Q0    = (const float*)d_in[2];
    const float* occ   = (const float*)d_in[3];
    const float* ext   = (const float*)d_in[4];
    const float* trans = (const float*)d_in[5];
    float* out = (float*)d_out;
    float* ws  = (float*)d_ws;   // needs NBLOCKS*WS_STRIDE*4 = 16 KB

    ct_stage1<<<NBLOCKS, NTHREADS, 0, stream>>>(rots, P0, Q0, occ, ext, trans, ws);
    ct_stage2<<<1, 32, 0, stream>>>(ws, out);
}